// MSDeformAttn_with_GlobalRegisters_71279277244641
// MI455X (gfx1250) — compile-verified
//
#include <hip/hip_runtime.h>
#include <hip/hip_bf16.h>

// ---------------- problem constants (from reference) ----------------
constexpr int kB    = 8;
constexpr int kNH   = 8;
constexpr int kHD   = 32;     // == wave32 lanes, one lane per head channel
constexpr int kL    = 4;
constexpr int kP    = 4;
constexpr int kR    = 16;
constexpr int kD    = 256;
constexpr int kHW   = 5440;   // 64*64 + 32*32 + 16*16 + 8*8
constexpr int kLEN  = 5456;   // kHW + kR
constexpr int kTOK  = kB * kLEN;       // 43648 total tokens
constexpr int kMT   = kTOK / 16;       // 2728 M-tiles of 16 rows
constexpr int kGX   = kMT / 8;         // 341 blocks of 8 waves

typedef __attribute__((ext_vector_type(2))) float v2f;
typedef __attribute__((ext_vector_type(8))) float v8f;

// =====================================================================
// GEMM  C[kTOK x Nout] = A[kTOK x 256] * W[256 x Nout] + bias
// f32 WMMA 16x16x4. One workgroup = 8 waves sharing one 256x64 weight
// panel staged in LDS (64 KB of the 320 KB/WGP); each wave owns a
// 16-row M-tile and computes a 16x64 output strip.
//
// LDS panel is K-pair interleaved:  Bs[(row>>1)*128 + col*2 + (row&1)]
// so a B fragment {W[2r][c], W[2r+1][c]} is one aligned ds_load_b64
// directly into an even VGPR pair (no repack movs before v_wmma).
//
// Fragment layout (ISA 7.12.2):
//   A 16x4 : lane&15 = M row; khalf=(lane>>4)*2 ; v0=K+khalf, v1=K+khalf+1
//   B 4x16 : lane&15 = N col; v0=row khalf, v1=row khalf+1
//   C 16x16: VGPR j -> M = j + (lane>>4)*8 ; N = lane&15
// =====================================================================
__global__ __launch_bounds__(256)
void gemm256_wmma(const float* __restrict__ A, const float* __restrict__ W,
                  const float* __restrict__ bias, float* __restrict__ C,
                  int Nout) {
  __shared__ float Bs[128 * 128];               // 128 K-pairs x (64 cols x 2) = 64 KB

  const int lane  = threadIdx.x & 31;
  const int wave  = threadIdx.x >> 5;
  const int mtile = blockIdx.x * 8 + wave;      // 16 output rows per wave
  const int ncol  = blockIdx.y * 64;            // 64 output cols per block
  const int colL  = lane & 15;
  const int khalf = (lane >> 4) << 1;           // 0 or 2

  // ---- cooperative stage of the weight panel into LDS (K-pair interleaved)
  // 4096 float4 micro-tiles (2 rows x 2 cols) / 256 threads = 16 per thread
  #pragma unroll
  for (int i = threadIdx.x; i < 4096; i += 256) {
    const int r = i >> 5;                       // K pair 0..127
    const int c = (i & 31) << 1;                // col 0,2,...,62
    const float* __restrict__ g0 = W + (size_t)(2 * r)     * Nout + ncol + c;
    const float* __restrict__ g1 = W + (size_t)(2 * r + 1) * Nout + ncol + c;
    float4 v;
    v.x = g0[0]; v.y = g1[0]; v.z = g0[1]; v.w = g1[1];
    *(float4*)&Bs[r * 128 + c * 2] = v;
  }
  __syncthreads();                              // s_barrier_signal/_wait

  const float* __restrict__ arow = A + (size_t)(mtile * 16 + colL) * kD;

  v8f acc0 = {}, acc1 = {}, acc2 = {}, acc3 = {};
  for (int k = 0; k < kD; k += 4) {
    __builtin_prefetch(arow + k + 32, 0, 3);    // global_prefetch_b8 (near)
    v2f a = *(const v2f*)(arow + k + khalf);
    const float* __restrict__ w0 = &Bs[((k + khalf) >> 1) * 128 + colL * 2];
    v2f b0 = *(const v2f*)(w0);                 // ds_load_b64: rows khalf,khalf+1
    v2f b1 = *(const v2f*)(w0 + 32);            // cols +16
    v2f b2 = *(const v2f*)(w0 + 64);            // cols +32
    v2f b3 = *(const v2f*)(w0 + 96);            // cols +48
    acc0 = __builtin_amdgcn_wmma_f32_16x16x4_f32(false, a, false, b0, (short)0, acc0, false, false);
    acc1 = __builtin_amdgcn_wmma_f32_16x16x4_f32(false, a, false, b1, (short)0, acc1, false, false);
    acc2 = __builtin_amdgcn_wmma_f32_16x16x4_f32(false, a, false, b2, (short)0, acc2, false, false);
    acc3 = __builtin_amdgcn_wmma_f32_16x16x4_f32(false, a, false, b3, (short)0, acc3, false, false);
  }

  const float bb0 = bias[ncol + colL];
  const float bb1 = bias[ncol + colL + 16];
  const float bb2 = bias[ncol + colL + 32];
  const float bb3 = bias[ncol + colL + 48];
  const int rbase = mtile * 16 + ((lane >> 4) << 3);
  #pragma unroll
  for (int j = 0; j < 8; ++j) {
    float* __restrict__ crow = C + (size_t)(rbase + j) * Nout + ncol + colL;
    crow[0]  = acc0[j] + bb0;
    crow[16] = acc1[j] + bb1;
    crow[32] = acc2[j] + bb2;
    crow[48] = acc3[j] + bb3;
  }
}

// =====================================================================
// In-place softmax over rows of 16 (aw: L*P=16, reg_w: R=16)
// =====================================================================
__global__ __launch_bounds__(256)
void softmax16(float* __restrict__ p, int rows) {
  const int r = blockIdx.x * 256 + threadIdx.x;
  if (r >= rows) return;
  float* __restrict__ x = p + (size_t)r * 16;
  float m = x[0];
  #pragma unroll
  for (int i = 1; i < 16; ++i) m = fmaxf(m, x[i]);
  float e[16];
  float s = 0.f;
  #pragma unroll
  for (int i = 0; i < 16; ++i) { e[i] = expf(x[i] - m); s += e[i]; }
  const float inv = 1.f / s;
  #pragma unroll
  for (int i = 0; i < 16; ++i) x[i] = e[i] * inv;
}

// =====================================================================
// Fused deformable bilinear sampling + register-token attention.
// One wave per (b, lq, h); lane = head channel (HD == 32 == wave32).
// value layout: (B, LEN_IN, NH, HD) row-major == (B, LEN_IN, 256)
//   tokens [0, R)       -> register values
//   tokens [R, R+HW)    -> spatial values (level l at R + start[l])
// out: combined hw_key + reg_key, (B, LEN_IN, 256)
// =====================================================================
__global__ __launch_bounds__(256)
void deform_combine(const float* __restrict__ value,
                    const float* __restrict__ refpts,   // (B, LEN_IN, L, 2)
                    const float* __restrict__ offs,     // (B, LEN_IN, 256) = (NH,L,P,2)
                    const float* __restrict__ aw,       // (B, LEN_IN, NH, 16) softmaxed
                    const float* __restrict__ rgw,      // (B, LEN_IN, NH, 16) softmaxed
                    float* __restrict__ out) {
  const int gw   = (blockIdx.x * 256 + (int)threadIdx.x) >> 5;
  const int lane = threadIdx.x & 31;
  if (gw >= kB * kLEN * kNH) return;
  const int h  = gw % kNH;
  const int lq = (gw / kNH) % kLEN;
  const int b  = gw / (kNH * kLEN);

  const int Hs[kL] = {64, 32, 16, 8};
  const int Ws[kL] = {64, 32, 16, 8};
  const int St[kL] = {0, 4096, 5120, 5376};

  const size_t tok = (size_t)b * kLEN + lq;
  const float* __restrict__ vcol = value + (size_t)b * kLEN * kD + h * kHD + lane;
  const float* __restrict__ wrow = rgw + (tok * kNH + h) * 16;
  const float* __restrict__ arow = aw  + (tok * kNH + h) * 16;
  const float* __restrict__ orow = offs + tok * kD + h * (kL * kP * 2);
  const float* __restrict__ rrow = refpts + tok * (kL * 2);

  // register-token attention: sum_r reg_w[r] * value[b, r, h, lane]
  float acc = 0.f;
  #pragma unroll
  for (int r = 0; r < kR; ++r) acc += wrow[r] * vcol[(size_t)r * kD];

  // multi-scale deformable sampling (zero-pad bilinear, align_corners=False)
  for (int l = 0; l < kL; ++l) {
    const int Hi = Hs[l], Wi = Ws[l];
    const float fH = (float)Hi, fW = (float)Wi;
    const float rx = rrow[l * 2 + 0];
    const float ry = rrow[l * 2 + 1];
    const float* __restrict__ vlev = vcol + (size_t)(kR + St[l]) * kD;
    #pragma unroll
    for (int p = 0; p < kP; ++p) {
      const float ox = orow[(l * kP + p) * 2 + 0];
      const float oy = orow[(l * kP + p) * 2 + 1];
      const float lx = rx + ox / fW;                // loc in [0,1]
      const float ly = ry + oy / fH;
      const float x = lx * fW - 0.5f;
      const float y = ly * fH - 0.5f;
      const float x0f = floorf(x), y0f = floorf(y);
      const int ix0 = (int)x0f, iy0 = (int)y0f;
      const float wx1 = x - x0f, wx0 = 1.f - wx1;
      const float wy1 = y - y0f, wy0 = 1.f - wy1;
      float s = 0.f;
      if (ix0 >= 0     && ix0 < Wi     && iy0 >= 0     && iy0 < Hi)
        s += wx0 * wy0 * vlev[(size_t)(iy0 * Wi + ix0) * kD];
      if (ix0 + 1 >= 0 && ix0 + 1 < Wi && iy0 >= 0     && iy0 < Hi)
        s += wx1 * wy0 * vlev[(size_t)(iy0 * Wi + ix0 + 1) * kD];
      if (ix0 >= 0     && ix0 < Wi     && iy0 + 1 >= 0 && iy0 + 1 < Hi)
        s += wx0 * wy1 * vlev[(size_t)((iy0 + 1) * Wi + ix0) * kD];
      if (ix0 + 1 >= 0 && ix0 + 1 < Wi && iy0 + 1 >= 0 && iy0 + 1 < Hi)
        s += wx1 * wy1 * vlev[(size_t)((iy0 + 1) * Wi + ix0 + 1) * kD];
      acc += arow[l * kP + p] * s;
    }
  }

  out[tok * kD + h * kHD + lane] = acc;   // LAM = (1,1): hw_key + reg_key
}

// =====================================================================
extern "C" void kernel_launch(void* const* d_in, const int* in_sizes, int n_in,
                              void* d_out, int out_size, void* d_ws, size_t ws_size,
                              hipStream_t stream) {
  const float* query  = (const float*)d_in[0];
  const float* refpts = (const float*)d_in[1];
  const float* inflat = (const float*)d_in[2];
  // d_in[3] shapes, d_in[4] starts, d_in[5] hw_sigma: compile-time constants here
  const float* Wv   = (const float*)d_in[6];
  const float* bv   = (const float*)d_in[7];
  const float* Wo   = (const float*)d_in[8];
  const float* bo   = (const float*)d_in[9];
  const float* Wa   = (const float*)d_in[10];
  const float* ba   = (const float*)d_in[11];
  const float* Wr   = (const float*)d_in[12];
  const float* br   = (const float*)d_in[13];
  const float* Wout = (const float*)d_in[14];
  const float* bout = (const float*)d_in[15];
  float* out = (float*)d_out;

  // workspace layout (f32): value | offsets | aw | reg_w | combined  (~179 MB)
  float* ws      = (float*)d_ws;
  float* value   = ws;                               // kTOK*256
  float* offsets = value   + (size_t)kTOK * 256;     // kTOK*256
  float* awb     = offsets + (size_t)kTOK * 256;     // kTOK*128
  float* rgwb    = awb     + (size_t)kTOK * 128;     // kTOK*128
  float* comb    = rgwb    + (size_t)kTOK * 128;     // kTOK*256

  // 1) projections (WMMA GEMMs, K=256, LDS-staged weight panels)
  gemm256_wmma<<<dim3(kGX, 4), 256, 0, stream>>>(inflat, Wv, bv, value,   256);
  gemm256_wmma<<<dim3(kGX, 4), 256, 0, stream>>>(query,  Wo, bo, offsets, 256);
  gemm256_wmma<<<dim3(kGX, 2), 256, 0, stream>>>(query,  Wa, ba, awb,     128);
  gemm256_wmma<<<dim3(kGX, 2), 256, 0, stream>>>(query,  Wr, br, rgwb,    128);

  // 2) softmax over 16-wide rows
  const int rows = kTOK * kNH;                       // 349184
  softmax16<<<(rows + 255) / 256, 256, 0, stream>>>(awb,  rows);
  softmax16<<<(rows + 255) / 256, 256, 0, stream>>>(rgwb, rows);

  // 3) fused sampling + register attention: one wave per (b, lq, h)
  const int waves = kTOK * kNH;
  deform_combine<<<(waves * 32 + 255) / 256, 256, 0, stream>>>(
      value, refpts, offsets, awb, rgwb, comb);

  // 4) output projection
  gemm256_wmma<<<dim3(kGX, 4), 256, 0, stream>>>(comb, Wout, bout, out, 256);
}